// HyperFormerConv_45621142618218
// MI455X (gfx1250) — compile-verified
//
#include <hip/hip_runtime.h>
#include <hip/hip_bf16.h>
#include <math.h>

// ---------------- problem constants (from reference) ----------------
#define N_NODES 4096
#define DIN     64
#define DOUT    32
#define NHEAD   4
#define EPSF    1e-7f
// C = 1, K = 1, TEMP = 1  -> all sqrt(K)/c factors are 1.

typedef __attribute__((ext_vector_type(2))) float v2f;
typedef __attribute__((ext_vector_type(8))) float v8f;

// ---------------- WMMA f32 16x16x4 wrapper ----------------
// A: 16x4 f32 (2 VGPR/lane), B: 4x16 f32 (2 VGPR/lane), C/D: 16x16 f32 (8 VGPR/lane).
// A layout: row = lane&15, a.x=A[row][2g], a.y=A[row][2g+1], g=lane>>4 (per 4-wide K chunk)
// B layout: col = lane&15, b.x=B[2g][col], b.y=B[2g+1][col]
// D layout: element (M,N) -> vgpr M&7, lane = (M>=8?16:0) + N
__device__ __forceinline__ v8f wmma_f32x4(v2f a, v2f b, v8f c) {
#if defined(__gfx1250__) && __has_builtin(__builtin_amdgcn_wmma_f32_16x16x4_f32)
    return __builtin_amdgcn_wmma_f32_16x16x4_f32(false, a, false, b, (short)0, c,
                                                 false, false);
#else
    // placeholder fallback (never correct, only keeps non-gfx1250 compiles alive)
    c[0] += a.x * b.x + a.y * b.y;
    return c;
#endif
}

__device__ __forceinline__ float halfgrp_sum(float v) {
    // reduce across the 16-lane half-group (masks < 16 stay inside the group)
    v += __shfl_xor(v, 1);
    v += __shfl_xor(v, 2);
    v += __shfl_xor(v, 4);
    v += __shfl_xor(v, 8);
    return v;
}
__device__ __forceinline__ float halfgrp_max(float v) {
    v = fmaxf(v, __shfl_xor(v, 1));
    v = fmaxf(v, __shfl_xor(v, 2));
    v = fmaxf(v, __shfl_xor(v, 4));
    v = fmaxf(v, __shfl_xor(v, 8));
    return v;
}

// ---------------- kernel 1: logmap0 (per node, 64-d) ----------------
__global__ void k_logmap0(const float* __restrict__ x, float* __restrict__ xt, int n) {
    int idx = blockIdx.x * blockDim.x + threadIdx.x;
    if (idx >= n) return;
    const float* r = x + (size_t)idx * DIN;
    float ss = 0.f;
#pragma unroll
    for (int d = 1; d < DIN; ++d) { float v = r[d]; ss += v * v; }
    float dn   = fmaxf(sqrtf(ss), EPSF);
    float x0   = fmaxf(r[0], 1.0f + EPSF);           // sqrt(K) = 1
    float coef = acoshf(x0) / dn;
    float* o = xt + (size_t)idx * DIN;
    o[0] = 0.f;
#pragma unroll
    for (int d = 1; d < DIN; ++d) o[d] = coef * r[d];
}

// ---------------- kernel 2: hyp_linear (WMMA GEMM + expmap0 epilogue) ----------------
struct LinArgs {
    const float* xt[2];   // log-mapped u, i     [N x 64]
    const float* W[3];    // wq, wk, wv          [H x 32 x 64]
    const float* B[3];    // bq, bk, bv          [H x 32]
    float*       out[6];  // Qu,Ku,Vu,Qi,Ki,Vi   [H x N x 32]
};

__global__ void __launch_bounds__(32) k_hyp_linear(LinArgs args) {
    const int lane = threadIdx.x;
    const int col  = lane & 15;       // also the A-row index
    const int g    = lane >> 4;
    const int tile = blockIdx.x;      // N/16
    const int h    = blockIdx.y;      // head
    const int p    = blockIdx.z;      // 0..5 = ui*3 + proj
    const int ui = p / 3, proj = p % 3;

    const float* xt = args.xt[ui] + (size_t)tile * 16 * DIN;
    const float* W  = args.W[proj] + (size_t)h * DOUT * DIN;
    const float* Bb = args.B[proj] + (size_t)h * DOUT;
    float* out = args.out[p] + ((size_t)h * N_NODES + (size_t)tile * 16) * DOUT;

    v8f acc0 = {}; v8f acc1 = {};
#pragma unroll
    for (int k0 = 0; k0 < DIN; k0 += 4) {
        const int kb = k0 + 2 * g;
        v2f a;  a.x  = xt[col * DIN + kb];          a.y  = xt[col * DIN + kb + 1];
        v2f b0; b0.x = W[col * DIN + kb];           b0.y = W[col * DIN + kb + 1];
        v2f b1; b1.x = W[(col + 16) * DIN + kb];    b1.y = W[(col + 16) * DIN + kb + 1];
        acc0 = wmma_f32x4(a, b0, acc0);
        acc1 = wmma_f32x4(a, b1, acc1);
    }

    // epilogue: mv[...,0]=0, +bias (bias[...,0]=0), expmap0
    const float bv0 = (col == 0) ? 0.f : Bb[col];
    const float bv1 = Bb[col + 16];
#pragma unroll
    for (int r = 0; r < 8; ++r) {
        float v0 = ((col == 0) ? 0.f : acc0[r]) + bv0;   // component col
        float v1 = acc1[r] + bv1;                        // component col+16
        float part = ((col == 0) ? 0.f : v0 * v0) + v1 * v1;  // exclude comp 0
        part = halfgrp_sum(part);
        float dn = fmaxf(sqrtf(part), EPSF);             // theta (sqrtK=1)
        float sh = sinhf(dn) / dn;
        float o0 = (col == 0) ? coshf(dn) : sh * v0;
        float o1 = sh * v1;
        int rr = r + 8 * g;                              // D-layout row
        out[rr * DOUT + col]      = o0;
        out[rr * DOUT + col + 16] = o1;
    }
}

// ---------------- kernel 3: hyperbolic flash attention ----------------
struct AttnArgs {
    const float* Q[2];
    const float* K[2];
    const float* V[2];
    float*       out[2];
};

__global__ void __launch_bounds__(32) k_attention(AttnArgs args) {
    __shared__ float Pt[16][17];   // P tile, padded to dodge bank conflicts

    const int lane = threadIdx.x;
    const int col  = lane & 15;
    const int g    = lane >> 4;
    const int qt = blockIdx.x, h = blockIdx.y, z = blockIdx.z;

    const float* Q  = args.Q[z] + ((size_t)h * N_NODES + (size_t)qt * 16) * DOUT;
    const float* Kp = args.K[z] + (size_t)h * N_NODES * DOUT;
    const float* Vp = args.V[z] + (size_t)h * N_NODES * DOUT;
    float* out = args.out[z] + ((size_t)h * N_NODES + (size_t)qt * 16) * DOUT;

    // pre-stage Q fragments in A layout: qa[t] covers K-chunk [4t, 4t+4)
    v2f qa[8];
#pragma unroll
    for (int t = 0; t < 8; ++t) {
        qa[t].x = Q[col * DOUT + 4 * t + 2 * g];
        qa[t].y = Q[col * DOUT + 4 * t + 2 * g + 1];
    }

    v8f acc0 = {}, acc1 = {};
    float mrow[8], ssum[8];
#pragma unroll
    for (int r = 0; r < 8; ++r) { mrow[r] = -3.0e38f; ssum[r] = 0.f; }

    const int NT = N_NODES / 16;
    for (int kt = 0; kt < NT; ++kt) {
        const float* Kb = Kp + (size_t)kt * 16 * DOUT;
        const float* Vb = Vp + (size_t)kt * 16 * DOUT;
        __builtin_prefetch(Kb + 16 * DOUT, 0, 0);   // next K tile -> global_prefetch
        __builtin_prefetch(Vb + 16 * DOUT, 0, 0);   // next V tile

        // S = Q . K'^T, with k'_0 = -k_0  (Minkowski dot folded into GEMM)
        v8f s = {};
#pragma unroll
        for (int t = 0; t < 8; ++t) {
            const int kb = 4 * t + 2 * g;
            v2f b;
            b.x = Kb[col * DOUT + kb];
            b.y = Kb[col * DOUT + kb + 1];
            if (t == 0) b.x = (g == 0) ? -b.x : b.x;   // component 0 negated
            s = wmma_f32x4(qa[t], b, s);
        }

        // logits, streaming softmax stats (rows live in 16-lane half-groups)
        float pv[8], fac[8];
#pragma unroll
        for (int r = 0; r < 8; ++r) {
            float mink  = s[r];
            float xx    = fmaxf(-mink, 1.0f + EPSF);    // K = 1
            float ac    = acoshf(xx);
            float logit = -(ac * ac);                   // TEMP = 1
            float mx    = halfgrp_max(logit);
            float mnew  = fmaxf(mrow[r], mx);
            float f     = __expf(mrow[r] - mnew);
            float p     = __expf(logit - mnew);
            float sp    = halfgrp_sum(p);
            ssum[r] = ssum[r] * f + sp;
            mrow[r] = mnew;
            fac[r]  = f;
            pv[r]   = p;
        }
#pragma unroll
        for (int r = 0; r < 8; ++r) { acc0[r] *= fac[r]; acc1[r] *= fac[r]; }

        // D-layout -> A-layout for P via LDS
        __syncthreads();
#pragma unroll
        for (int r = 0; r < 8; ++r) Pt[r + 8 * g][col] = pv[r];
        __syncthreads();

        // acc += P (16x16) . V (16x32)
#pragma unroll
        for (int l0 = 0; l0 < 16; l0 += 4) {
            const int kr = l0 + 2 * g;
            v2f a;  a.x  = Pt[col][kr];               a.y  = Pt[col][kr + 1];
            v2f b0; b0.x = Vb[kr * DOUT + col];       b0.y = Vb[(kr + 1) * DOUT + col];
            v2f b1; b1.x = Vb[kr * DOUT + col + 16];  b1.y = Vb[(kr + 1) * DOUT + col + 16];
            acc0 = wmma_f32x4(a, b0, acc0);
            acc1 = wmma_f32x4(a, b1, acc1);
        }
    }

    // finalize: softmax normalize, Minkowski-norm rescale
#pragma unroll
    for (int r = 0; r < 8; ++r) {
        float inv = 1.0f / ssum[r];
        float a0 = acc0[r] * inv;
        float a1 = acc1[r] * inv;
        float part = ((col == 0) ? -a0 * a0 : a0 * a0) + a1 * a1;  // mink_norm
        part = halfgrp_sum(part);
        float coef = rsqrtf(fabsf(part) + EPSF);                    // c = 1
        int rr = r + 8 * g;
        out[rr * DOUT + col]      = a0 * coef;
        out[rr * DOUT + col + 16] = a1 * coef;
    }
}

// ---------------- kernel 4: mobius_add (per (h,n) row) ----------------
__global__ void k_mobius(const float* __restrict__ x, const float* __restrict__ y,
                         float* __restrict__ o, int nrows) {
    int idx = blockIdx.x * blockDim.x + threadIdx.x;
    if (idx >= nrows) return;
    const float* xr = x + (size_t)idx * DOUT;
    const float* yr = y + (size_t)idx * DOUT;
    float xy = 0.f, x2 = 0.f, y2 = 0.f;
#pragma unroll
    for (int d = 0; d < DOUT; ++d) {
        float xv = xr[d], yv = yr[d];
        xy += xv * yv; x2 += xv * xv; y2 += yv * yv;
    }
    float ca  = 1.0f + 2.0f * xy + y2;     // c = 1
    float cb  = 1.0f - x2;
    float den = 1.0f + 2.0f * xy + x2 * y2 + EPSF;
    float inv = 1.0f / den;
    float* orow = o + (size_t)idx * DOUT;
#pragma unroll
    for (int d = 0; d < DOUT; ++d) orow[d] = (ca * xr[d] + cb * yr[d]) * inv;
}

// ---------------- kernel 5: hyper_agg (mean over heads + rescale) ----------------
__global__ void k_agg(const float* __restrict__ comb, float* __restrict__ out, int n) {
    int idx = blockIdx.x * blockDim.x + threadIdx.x;
    if (idx >= n) return;
    float s[DOUT];
#pragma unroll
    for (int d = 0; d < DOUT; ++d) {
        float acc = 0.f;
#pragma unroll
        for (int h = 0; h < NHEAD; ++h)
            acc += comb[((size_t)h * N_NODES + idx) * DOUT + d];
        s[d] = acc * (1.0f / NHEAD);
    }
    float mn = -s[0] * s[0];
#pragma unroll
    for (int d = 1; d < DOUT; ++d) mn += s[d] * s[d];
    float coef = rsqrtf(fabsf(mn) + EPSF);   // c = 1
    float* orow = out + (size_t)idx * DOUT;
#pragma unroll
    for (int d = 0; d < DOUT; ++d) orow[d] = s[d] * coef;
}

// ---------------- host orchestration ----------------
extern "C" void kernel_launch(void* const* d_in, const int* in_sizes, int n_in,
                              void* d_out, int out_size, void* d_ws, size_t ws_size,
                              hipStream_t stream) {
    // setup_inputs order: u, i, adjs(unused), wq, bq, wk, bk, wv, bv
    const float* u  = (const float*)d_in[0];
    const float* iN = (const float*)d_in[1];
    const float* wq = (const float*)d_in[3];
    const float* bq = (const float*)d_in[4];
    const float* wk = (const float*)d_in[5];
    const float* bk = (const float*)d_in[6];
    const float* wv = (const float*)d_in[7];
    const float* bv = (const float*)d_in[8];
    float* out = (float*)d_out;

    float* w = (float*)d_ws;
    size_t off = 0;
    auto alloc = [&](size_t nflt) { float* p = w + off; off += nflt; return p; };
    const size_t NQKV = (size_t)NHEAD * N_NODES * DOUT;

    float* xt_u = alloc((size_t)N_NODES * DIN);
    float* xt_i = alloc((size_t)N_NODES * DIN);
    float* Qu = alloc(NQKV); float* Ku = alloc(NQKV); float* Vu = alloc(NQKV);
    float* Qi = alloc(NQKV); float* Ki = alloc(NQKV); float* Vi = alloc(NQKV);
    float* av_u = alloc(NQKV); float* av_i = alloc(NQKV);   // hyper_softmax outputs
    float* cb_u = alloc(NQKV); float* cb_i = alloc(NQKV);   // mobius outputs
    (void)ws_size; (void)in_sizes; (void)n_in; (void)out_size;

    k_logmap0<<<dim3((N_NODES + 255) / 256), dim3(256), 0, stream>>>(u,  xt_u, N_NODES);
    k_logmap0<<<dim3((N_NODES + 255) / 256), dim3(256), 0, stream>>>(iN, xt_i, N_NODES);

    LinArgs la;
    la.xt[0] = xt_u; la.xt[1] = xt_i;
    la.W[0] = wq; la.W[1] = wk; la.W[2] = wv;
    la.B[0] = bq; la.B[1] = bk; la.B[2] = bv;
    la.out[0] = Qu; la.out[1] = Ku; la.out[2] = Vu;
    la.out[3] = Qi; la.out[4] = Ki; la.out[5] = Vi;
    k_hyp_linear<<<dim3(N_NODES / 16, NHEAD, 6), dim3(32), 0, stream>>>(la);

    AttnArgs aa;
    aa.Q[0] = Qu; aa.K[0] = Ki; aa.V[0] = Vi; aa.out[0] = av_u;  // softmax(u_q, i_k, i_v)
    aa.Q[1] = Qi; aa.K[1] = Ku; aa.V[1] = Vu; aa.out[1] = av_i;  // softmax(i_q, u_k, u_v)
    k_attention<<<dim3(N_NODES / 16, NHEAD, 2), dim3(32), 0, stream>>>(aa);

    const int HR = NHEAD * N_NODES;
    k_mobius<<<dim3((HR + 255) / 256), dim3(256), 0, stream>>>(Vu, av_u, cb_u, HR);
    k_mobius<<<dim3((HR + 255) / 256), dim3(256), 0, stream>>>(Vi, av_i, cb_i, HR);

    k_agg<<<dim3((N_NODES + 255) / 256), dim3(256), 0, stream>>>(cb_u, out, N_NODES);
    k_agg<<<dim3((N_NODES + 255) / 256), dim3(256), 0, stream>>>(
        cb_i, out + (size_t)N_NODES * DOUT, N_NODES);
}